// CustomGNNModel_74002286510429
// MI455X (gfx1250) — compile-verified
//
#include <hip/hip_runtime.h>
#include <hip/hip_bf16.h>

typedef __attribute__((ext_vector_type(2))) float v2f;
typedef __attribute__((ext_vector_type(8))) float v8f;

#define NNODES 100000
#define NEDGES 3200000
#define INDIM  512
#define HID    16
#define ODIM   40
#define OPAD   48
#define NTILES (NNODES / 16)   // 6250, exact

// ---------------- degree / normalization ----------------

__global__ void k_deg_init(float* __restrict__ deg) {
    int v = blockIdx.x * blockDim.x + threadIdx.x;
    if (v < NNODES) deg[v] = 1.0f;   // self loop
}

__global__ void k_deg_count(const int* __restrict__ ei, float* __restrict__ deg) {
    int e = blockIdx.x * blockDim.x + threadIdx.x;
    if (e < NEDGES) {
        int d = ei[NEDGES + e];      // dst row of edge_index
        atomicAdd(&deg[d], 1.0f);
    }
}

__global__ void k_dinv(float* __restrict__ deg) {
    int v = blockIdx.x * blockDim.x + threadIdx.x;
    if (v < NNODES) deg[v] = rsqrtf(deg[v]);
}

// ---------------- GEMM1: h1 = x @ W1  (100000x512 @ 512x16) ----------------
// One wave per 16-row tile. W1 staged in LDS (32 KB). K-loop of 128
// v_wmma_f32_16x16x4_f32 accumulations.

__global__ __launch_bounds__(256) void k_gemm1(const float* __restrict__ x,
                                               const float* __restrict__ W1,
                                               float* __restrict__ h1) {
    __shared__ float sW[INDIM * HID];           // 32 KB
    for (int i = threadIdx.x; i < INDIM * HID; i += 256) sW[i] = W1[i];
    __syncthreads();

    int wave = threadIdx.x >> 5;
    int lane = threadIdx.x & 31;
    int tile = blockIdx.x * 8 + wave;
    if (tile >= NTILES) return;                 // wave-uniform guard

    int n    = lane & 15;                       // A row / B col within tile
    int koff = (lane >> 4) << 1;                // lanes 16-31 hold K+2,K+3
    const float* xr = x + (size_t)(tile * 16 + n) * INDIM + koff;

    v8f c = {};
#pragma unroll 4
    for (int k0 = 0; k0 < INDIM; k0 += 4) {
        v2f a = *(const v2f*)(xr + k0);         // A[m=n][k0+koff, k0+koff+1]
        v2f b;
        b.x = sW[(k0 + koff) * HID + n];        // B[k0+koff][n]
        b.y = sW[(k0 + koff + 1) * HID + n];    // B[k0+koff+1][n]
        c = __builtin_amdgcn_wmma_f32_16x16x4_f32(false, a, false, b,
                                                  (short)0, c, false, false);
    }

    int mbase = tile * 16 + ((lane >> 4) << 3); // VGPR r: M=r (lanes 0-15), M=r+8 (16-31)
#pragma unroll
    for (int r = 0; r < 8; ++r)
        h1[(size_t)(mbase + r) * HID + n] = c[r];
}

// ---------------- layer-1 aggregation ----------------

__global__ void k_self16(const float* __restrict__ dinv, const float* __restrict__ h1,
                         float* __restrict__ agg) {
    int idx = blockIdx.x * blockDim.x + threadIdx.x;
    if (idx < NNODES * HID) {
        float w = dinv[idx >> 4];
        agg[idx] = w * w * h1[idx];
    }
}

__global__ void k_edges16(const int* __restrict__ ei, const float* __restrict__ dinv,
                          const float* __restrict__ h1, float* __restrict__ agg) {
    unsigned idx = blockIdx.x * blockDim.x + threadIdx.x;
    if (idx < (unsigned)NEDGES * HID) {
        int e = idx >> 4, cch = idx & 15;
        int s = ei[e], d = ei[NEDGES + e];
        float w = dinv[s] * dinv[d];
        atomicAdd(&agg[(size_t)d * HID + cch], w * h1[(size_t)s * HID + cch]);
    }
}

__global__ void k_relu_bias(float* __restrict__ agg, const float* __restrict__ b1) {
    int idx = blockIdx.x * blockDim.x + threadIdx.x;
    if (idx < NNODES * HID) {
        float v = agg[idx] + b1[idx & 15];
        agg[idx] = v > 0.0f ? v : 0.0f;
    }
}

// ---------------- GEMM2: h2 = relu_out @ W2  (100000x16 @ 16x40, padded to 48) ----

__global__ void k_padw2(const float* __restrict__ W2, float* __restrict__ W2p) {
    int i = blockIdx.x * blockDim.x + threadIdx.x;
    if (i < HID * OPAD) {
        int k = i / OPAD, nn = i % OPAD;
        W2p[i] = (nn < ODIM) ? W2[k * ODIM + nn] : 0.0f;
    }
}

__global__ __launch_bounds__(256) void k_gemm2(const float* __restrict__ h,
                                               const float* __restrict__ W2p,
                                               float* __restrict__ h2) {
    __shared__ float sW[HID * OPAD];            // 3 KB
    for (int i = threadIdx.x; i < HID * OPAD; i += 256) sW[i] = W2p[i];
    __syncthreads();

    int wave = threadIdx.x >> 5;
    int lane = threadIdx.x & 31;
    int tile = blockIdx.x * 8 + wave;
    if (tile >= NTILES) return;

    int n    = lane & 15;
    int koff = (lane >> 4) << 1;
    const float* hr = h + (size_t)(tile * 16 + n) * HID + koff;
    v2f a0 = *(const v2f*)(hr + 0);
    v2f a1 = *(const v2f*)(hr + 4);
    v2f a2 = *(const v2f*)(hr + 8);
    v2f a3 = *(const v2f*)(hr + 12);

    int mbase = tile * 16 + ((lane >> 4) << 3);
#pragma unroll
    for (int nt = 0; nt < 3; ++nt) {
        int col = nt * 16 + n;
        v8f c = {};
        v2f b;
        b.x = sW[(0 + koff) * OPAD + col];  b.y = sW[(1 + koff) * OPAD + col];
        c = __builtin_amdgcn_wmma_f32_16x16x4_f32(false, a0, false, b, (short)0, c, false, false);
        b.x = sW[(4 + koff) * OPAD + col];  b.y = sW[(5 + koff) * OPAD + col];
        c = __builtin_amdgcn_wmma_f32_16x16x4_f32(false, a1, false, b, (short)0, c, false, false);
        b.x = sW[(8 + koff) * OPAD + col];  b.y = sW[(9 + koff) * OPAD + col];
        c = __builtin_amdgcn_wmma_f32_16x16x4_f32(false, a2, false, b, (short)0, c, false, false);
        b.x = sW[(12 + koff) * OPAD + col]; b.y = sW[(13 + koff) * OPAD + col];
        c = __builtin_amdgcn_wmma_f32_16x16x4_f32(false, a3, false, b, (short)0, c, false, false);

        if (col < ODIM) {
#pragma unroll
            for (int r = 0; r < 8; ++r)
                h2[(size_t)(mbase + r) * ODIM + col] = c[r];
        }
    }
}

// ---------------- layer-2 aggregation ----------------

__global__ void k_self40(const float* __restrict__ dinv, const float* __restrict__ h2,
                         float* __restrict__ agg) {
    int idx = blockIdx.x * blockDim.x + threadIdx.x;
    if (idx < NNODES * ODIM) {
        float w = dinv[idx / ODIM];
        agg[idx] = w * w * h2[idx];
    }
}

__global__ void k_edges40(const int* __restrict__ ei, const float* __restrict__ dinv,
                          const float* __restrict__ h2, float* __restrict__ agg) {
    unsigned idx = blockIdx.x * blockDim.x + threadIdx.x;
    if (idx < (unsigned)NEDGES * ODIM) {
        int e = idx / ODIM, cch = idx - e * ODIM;
        int s = ei[e], d = ei[NEDGES + e];
        float w = dinv[s] * dinv[d];
        atomicAdd(&agg[(size_t)d * ODIM + cch], w * h2[(size_t)s * ODIM + cch]);
    }
}

// ---------------- bias + log_softmax ----------------

__global__ void k_lsm(const float* __restrict__ agg, const float* __restrict__ b2,
                      float* __restrict__ out) {
    int v = blockIdx.x * blockDim.x + threadIdx.x;
    if (v >= NNODES) return;
    const float* z = agg + (size_t)v * ODIM;
    float zz[ODIM];
    float m = -3.402823466e+38f;
#pragma unroll
    for (int c = 0; c < ODIM; ++c) { zz[c] = z[c] + b2[c]; m = fmaxf(m, zz[c]); }
    float s = 0.0f;
#pragma unroll
    for (int c = 0; c < ODIM; ++c) s += __expf(zz[c] - m);
    float lse = m + __logf(s);
    float* o = out + (size_t)v * ODIM;
#pragma unroll
    for (int c = 0; c < ODIM; ++c) o[c] = zz[c] - lse;
}

// ---------------- launcher ----------------

extern "C" void kernel_launch(void* const* d_in, const int* in_sizes, int n_in,
                              void* d_out, int out_size, void* d_ws, size_t ws_size,
                              hipStream_t stream) {
    const float* x  = (const float*)d_in[0];
    const int*   ei = (const int*)d_in[1];      // edge_index [2, E]
    const float* W1 = (const float*)d_in[2];
    const float* b1 = (const float*)d_in[3];
    const float* W2 = (const float*)d_in[4];
    const float* b2 = (const float*)d_in[5];
    float* out = (float*)d_out;

    float* ws   = (float*)d_ws;
    float* dinv = ws;                                  // N
    float* h1   = dinv + NNODES;                       // N*16
    float* agg1 = h1 + (size_t)NNODES * HID;           // N*16
    float* w2p  = agg1 + (size_t)NNODES * HID;         // 16*48
    float* h2   = w2p + HID * OPAD;                    // N*40
    float* agg2 = h2 + (size_t)NNODES * ODIM;          // N*40

    const int B = 256;
    k_deg_init<<<(NNODES + B - 1) / B, B, 0, stream>>>(dinv);
    k_deg_count<<<(NEDGES + B - 1) / B, B, 0, stream>>>(ei, dinv);
    k_dinv<<<(NNODES + B - 1) / B, B, 0, stream>>>(dinv);

    k_gemm1<<<(NTILES + 7) / 8, B, 0, stream>>>(x, W1, h1);

    k_self16<<<(NNODES * HID + B - 1) / B, B, 0, stream>>>(dinv, h1, agg1);
    k_edges16<<<((unsigned)NEDGES * HID + B - 1) / B, B, 0, stream>>>(ei, dinv, h1, agg1);
    k_relu_bias<<<(NNODES * HID + B - 1) / B, B, 0, stream>>>(agg1, b1);

    k_padw2<<<(HID * OPAD + B - 1) / B, B, 0, stream>>>(W2, w2p);
    k_gemm2<<<(NTILES + 7) / 8, B, 0, stream>>>(agg1, w2p, h2);

    k_self40<<<(NNODES * ODIM + B - 1) / B, B, 0, stream>>>(dinv, h2, agg2);
    k_edges40<<<((unsigned)NEDGES * ODIM + B - 1) / B, B, 0, stream>>>(ei, dinv, h2, agg2);

    k_lsm<<<(NNODES + B - 1) / B, B, 0, stream>>>(agg2, b2, out);
}